// CluePredictor_gcn_48017734369917
// MI455X (gfx1250) — compile-verified
//
#include <hip/hip_runtime.h>
#include <math.h>

// Problem constants
#define Bn   64
#define Ln   512
#define EMBn 400
#define Hn   256
#define NLn  3
#define En   512
#define MAXAn 32
#define KT   64   // k-chunk staged to LDS by the TDM in k_axh

typedef __attribute__((ext_vector_type(2))) float v2f;
typedef __attribute__((ext_vector_type(8))) float v8f;
typedef __attribute__((ext_vector_type(4))) unsigned int u32x4;
typedef __attribute__((ext_vector_type(8))) int i32x8;
typedef __attribute__((ext_vector_type(4))) int i32x4;

__device__ __forceinline__ v8f wmma4(v2f a, v2f b, v8f c) {
  // V_WMMA_F32_16X16X4_F32 : D = A(16x4,f32) * B(4x16,f32) + C(16x16,f32)
  return __builtin_amdgcn_wmma_f32_16x16x4_f32(false, a, false, b, (short)0, c, false, false);
}

// Issue a TDM 2D tile load: tile_h rows x tile_w f32 elements, row stride
// `stride` elements, from global `gaddr` into LDS byte offset `lds_addr`.
// D# layout per CDNA5 ISA 8.3/8.4 (groups 2/3 zero: 2D tensor, no iterate).
// This toolchain exposes the 6-arg builtin (extra i32x8 group + cpol).
__device__ __forceinline__ void tdm_load_2d(unsigned lds_addr, const void* gaddr,
                                            unsigned tile_w, unsigned tile_h,
                                            unsigned stride) {
  unsigned long long ga = (unsigned long long)(size_t)gaddr;
  u32x4 g0;
  g0[0] = 1u;                                             // count=1, user descriptor
  g0[1] = lds_addr;                                       // lds_addr (bytes)
  g0[2] = (unsigned)(ga & 0xFFFFFFFFu);                   // global_addr[31:0]
  g0[3] = (unsigned)((ga >> 32) & 0x01FFFFFFu) | (2u << 30); // addr[56:32] | type=2
  i32x8 g1;
  g1[0] = (2 << 16);                                      // wg_mask=0, data_size=2 (4B)
  g1[1] = (int)((tile_w & 0xFFFFu) << 16);                // tensor_dim0[15:0]
  g1[2] = (int)(((tile_w >> 16) & 0xFFFFu) |              // tensor_dim0[31:16]
                ((tile_h & 0xFFFFu) << 16));              // tensor_dim1[15:0]
  g1[3] = (int)(((tile_h >> 16) & 0xFFFFu) |              // tensor_dim1[31:16]
                ((tile_w & 0xFFFFu) << 16));              // tile_dim0
  g1[4] = (int)(tile_h & 0xFFFFu);                        // tile_dim1 (tile_dim2=0)
  g1[5] = (int)stride;                                    // tensor_dim0_stride[31:0]
  g1[6] = 0;                                              // stride0 hi, stride1 lo
  g1[7] = 0;
  i32x4 g2 = {0, 0, 0, 0};
  i32x4 g3 = {0, 0, 0, 0};
  i32x8 g4 = {0, 0, 0, 0, 0, 0, 0, 0};
  __builtin_amdgcn_tensor_load_to_lds(g0, g1, g2, g3, g4, 0);
}

// ---------------------------------------------------------------- utilities
__global__ void k_zero(float* __restrict__ p, int n) {
  int t = blockIdx.x * blockDim.x + threadIdx.x;
  for (int i = t; i < n; i += gridDim.x * blockDim.x) p[i] = 0.f;
}

// ---------------------------------------------------------------- 1x1 conv: X[b] = W1(HxEMB) * Semb[b](EMBxL)
__global__ void k_conv1(const float* __restrict__ W1, const float* __restrict__ Semb,
                        float* __restrict__ X) {
  const int b    = blockIdx.z;
  const int lane = threadIdx.x & 31;
  const int tid  = blockIdx.x * 8 + (threadIdx.x >> 5);
  const int nt   = Ln / 64;                       // 8 N-tiles
  const int m0 = (tid / nt) * 16, n0 = (tid % nt) * 64;
  const int r = lane & 15, hi = lane >> 4;
  const float* A  = W1 + (size_t)m0 * EMBn;
  const float* Bm = Semb + (size_t)b * EMBn * Ln + n0;
  float* C = X + (size_t)b * Hn * Ln;
  v8f acc[4] = {};
  for (int k = 0; k < EMBn; k += 4) {
    const float* ap = A + r * EMBn + k + 2 * hi;
    v2f a; a.x = ap[0]; a.y = ap[1];
#pragma unroll
    for (int j = 0; j < 4; ++j) {
      const float* bp = Bm + (k + 2 * hi) * Ln + j * 16 + r;
      v2f bv; bv.x = bp[0]; bv.y = bp[Ln];
      acc[j] = wmma4(a, bv, acc[j]);
    }
  }
#pragma unroll
  for (int j = 0; j < 4; ++j)
#pragma unroll
    for (int i = 0; i < 8; ++i)
      C[(m0 + i + 8 * hi) * Ln + n0 + j * 16 + r] = acc[j][i];
}

// ---------------------------------------------------------------- fused highway layer
__global__ void k_highway(const float* __restrict__ Wg, const float* __restrict__ bg,
                          const float* __restrict__ Wl, const float* __restrict__ bl,
                          const float* __restrict__ Xin, float* __restrict__ Xout) {
  const int b    = blockIdx.z;
  const int lane = threadIdx.x & 31;
  const int tid  = blockIdx.x * 8 + (threadIdx.x >> 5);
  const int nt   = Ln / 64;
  const int m0 = (tid / nt) * 16, n0 = (tid % nt) * 64;
  const int r = lane & 15, hi = lane >> 4;
  const float* xb = Xin + (size_t)b * Hn * Ln;
  float* ob = Xout + (size_t)b * Hn * Ln;
  v8f g[4] = {}, nlv[4] = {};
  for (int k = 0; k < Hn; k += 4) {
    const float* agp = Wg + (m0 + r) * Hn + k + 2 * hi;
    const float* alp = Wl + (m0 + r) * Hn + k + 2 * hi;
    v2f ag; ag.x = agp[0]; ag.y = agp[1];
    v2f al; al.x = alp[0]; al.y = alp[1];
#pragma unroll
    for (int j = 0; j < 4; ++j) {
      const float* bp = xb + (k + 2 * hi) * Ln + n0 + j * 16 + r;
      v2f bv; bv.x = bp[0]; bv.y = bp[Ln];
      g[j]   = wmma4(ag, bv, g[j]);
      nlv[j] = wmma4(al, bv, nlv[j]);
    }
  }
#pragma unroll
  for (int i = 0; i < 8; ++i) {
    const int m = m0 + i + 8 * hi;
    const float bgi = bg[m], bli = bl[m];
#pragma unroll
    for (int j = 0; j < 4; ++j) {
      const int col = n0 + j * 16 + r;
      const float xv = xb[m * Ln + col];
      const float gt = 1.f / (1.f + __expf(-(g[j][i] + bgi)));
      ob[m * Ln + col] = gt * (nlv[j][i] + bli) + (1.f - gt) * xv;
    }
  }
}

// ---------------------------------------------------------------- adjacency scatter (undirected, set-to-1 dedups)
__global__ void k_scatter(const int* __restrict__ edges, float* __restrict__ adj) {
  int t = blockIdx.x * blockDim.x + threadIdx.x;
  if (t >= Bn * En) return;
  int b = t / En;
  int s = edges[2 * t], d = edges[2 * t + 1];
  float* ab = adj + (size_t)b * Ln * Ln;
  ab[s * Ln + d] = 1.f;
  ab[d * Ln + s] = 1.f;
}

// ---------------------------------------------------------------- denom = rowsum(adj) + 1
__global__ void k_denom(const float* __restrict__ adj, float* __restrict__ den) {
  int t = blockIdx.x * blockDim.x + threadIdx.x;
  if (t >= Bn * Ln) return;
  const float4* row = (const float4*)(adj + (size_t)t * Ln);
  float s = 1.f;
  for (int i = 0; i < Ln / 4; ++i) { float4 v = row[i]; s += v.x + v.y + v.z + v.w; }
  den[t] = s;
}

// ---------------------------------------------------------------- X[B,H,L] -> H[B,L,H]
__global__ void k_transpose(const float* __restrict__ X, float* __restrict__ Hh) {
  __shared__ float tile[32][33];
  const int b = blockIdx.z;
  const int l0 = blockIdx.x * 32, c0 = blockIdx.y * 32;
  const int tx = threadIdx.x, ty = threadIdx.y;   // block (32,8)
  const float* xi = X + (size_t)b * Hn * Ln;
  float* ho = Hh + (size_t)b * Ln * Hn;
#pragma unroll
  for (int i = 0; i < 32; i += 8) tile[ty + i][tx] = xi[(c0 + ty + i) * Ln + l0 + tx];
  __syncthreads();
#pragma unroll
  for (int i = 0; i < 32; i += 8) ho[(l0 + ty + i) * Hn + c0 + tx] = tile[tx][ty + i];
}

// ---------------------------------------------------------------- Axh[b] = adj[b](LxL) * Hin[b](LxH) + Hin[b]
// Dominant GEMM. Block = 8 waves computing a 128(M) x 64(N) macro-tile.
// The shared B panel h[k0:k0+KT, n0:n0+64] is staged into LDS by the Tensor
// Data Mover, double-buffered: TDM for chunk c+1 overlaps WMMA on chunk c.
__global__ void k_axh(const float* __restrict__ adj, const float* __restrict__ Hin,
                      float* __restrict__ Axh) {
  __shared__ float bsm[2][KT * 64];               // 2 x 16 KB
  const int b    = blockIdx.z;
  const int lane = threadIdx.x & 31;
  const int w    = threadIdx.x >> 5;              // wave 0..7 -> M sub-tile
  const int nt   = Hn / 64;                       // 4 N macro-tiles
  const int m0 = (blockIdx.x / nt) * 128 + w * 16;
  const int n0 = (blockIdx.x % nt) * 64;
  const int r = lane & 15, hi = lane >> 4;
  const float* A  = adj + (size_t)b * Ln * Ln + (size_t)m0 * Ln;
  const float* hb = Hin + (size_t)b * Ln * Hn;
  float* O = Axh + (size_t)b * Ln * Hn;

  const unsigned lds0 = (unsigned)(size_t)(&bsm[0][0]);   // flat addr low 32b = LDS offset
  const unsigned lds1 = (unsigned)(size_t)(&bsm[1][0]);

  if (threadIdx.x == 0) {                         // TDM issue ignores EXEC; one wave issues
    tdm_load_2d(lds0, (const void*)(hb + n0), 64u, (unsigned)KT, (unsigned)Hn);
    __builtin_amdgcn_s_wait_tensorcnt(0);
  }
  __syncthreads();

  v8f acc[4] = {};
  for (int c = 0; c < Ln / KT; ++c) {
    const int p  = c & 1;
    const int k0 = c * KT;
    if (threadIdx.x == 0 && c + 1 < Ln / KT)      // prefetch next chunk into other buffer
      tdm_load_2d(p ? lds0 : lds1,
                  (const void*)(hb + (size_t)(k0 + KT) * Hn + n0),
                  64u, (unsigned)KT, (unsigned)Hn);
    if (c + 1 < Ln / KT)                          // stream next adj A-chunk toward caches
      __builtin_prefetch(A + r * Ln + k0 + KT, 0, 3);
    const float* bs = bsm[p];
    for (int kk = 0; kk < KT; kk += 4) {
      const float* ap = A + r * Ln + k0 + kk + 2 * hi;
      v2f a; a.x = ap[0]; a.y = ap[1];
#pragma unroll
      for (int j = 0; j < 4; ++j) {
        const float* bp = bs + (kk + 2 * hi) * 64 + j * 16 + r;
        v2f bv; bv.x = bp[0]; bv.y = bp[64];
        acc[j] = wmma4(a, bv, acc[j]);
      }
    }
    if (threadIdx.x == 0) __builtin_amdgcn_s_wait_tensorcnt(0);
    __syncthreads();                              // publish next buffer / retire this one
  }
#pragma unroll
  for (int j = 0; j < 4; ++j)
#pragma unroll
    for (int i = 0; i < 8; ++i) {
      const int m = m0 + i + 8 * hi, col = n0 + j * 16 + r;
      O[m * Hn + col] = acc[j][i] + hb[m * Hn + col];
    }
}

// ---------------------------------------------------------------- Hout = relu((Axh * W^T + 2b) / denom); W given row-major [O,H]
__global__ void k_gcnw(const float* __restrict__ Axh, const float* __restrict__ W,
                       const float* __restrict__ bias, const float* __restrict__ den,
                       float* __restrict__ Hout) {
  const int b    = blockIdx.z;
  const int lane = threadIdx.x & 31;
  const int tid  = blockIdx.x * 8 + (threadIdx.x >> 5);
  const int nt   = Hn / 64;
  const int m0 = (tid / nt) * 16, n0 = (tid % nt) * 64;
  const int r = lane & 15, hi = lane >> 4;
  const float* A  = Axh + (size_t)b * Ln * Hn + (size_t)m0 * Hn;
  const float* dn = den + b * Ln;
  float* O = Hout + (size_t)b * Ln * Hn;
  v8f acc[4] = {};
  for (int k = 0; k < Hn; k += 4) {
    const float* ap = A + r * Hn + k + 2 * hi;
    v2f a; a.x = ap[0]; a.y = ap[1];
#pragma unroll
    for (int j = 0; j < 4; ++j) {
      const float* bp = W + (n0 + j * 16 + r) * Hn + k + 2 * hi;  // B = W^T: contiguous float2
      v2f bv; bv.x = bp[0]; bv.y = bp[1];
      acc[j] = wmma4(a, bv, acc[j]);
    }
  }
#pragma unroll
  for (int j = 0; j < 4; ++j) {
    const int o = n0 + j * 16 + r;
    const float b2 = 2.f * bias[o];
#pragma unroll
    for (int i = 0; i < 8; ++i) {
      const int m = m0 + i + 8 * hi;
      const float v = (acc[j][i] + b2) / dn[m];
      O[m * Hn + o] = v > 0.f ? v : 0.f;
    }
  }
}

// ---------------------------------------------------------------- ypart[b,l] += dot(Hout[b,l,:], cw[:H])
__global__ void k_yacc(const float* __restrict__ Hout, const float* __restrict__ cw,
                       float* __restrict__ ypart) {
  int t = blockIdx.x * blockDim.x + threadIdx.x;
  if (t >= Bn * Ln) return;
  const float* hr = Hout + (size_t)t * Hn;
  float s = 0.f;
  for (int c = 0; c < Hn; ++c) s += hr[c] * cw[c];
  ypart[t] += s;
}

__global__ void k_yrelu(const float* __restrict__ ypart, const float* __restrict__ cb,
                        float* __restrict__ yrel) {
  int t = blockIdx.x * blockDim.x + threadIdx.x;
  if (t >= Bn * Ln) return;
  float v = ypart[t] + cb[0];
  yrel[t] = v > 0.f ? v : 0.f;
}

// ---------------------------------------------------------------- logits = mask(Y(BxL) * linW^T + lin_b)
__global__ void k_logits(const float* __restrict__ Y, const float* __restrict__ linW,
                         const float* __restrict__ linb, const int* __restrict__ wids,
                         float* __restrict__ out) {
  const int lane = threadIdx.x & 31;
  const int tid  = blockIdx.x * 8 + (threadIdx.x >> 5);
  const int nt   = Ln / 64;                       // 8; M=Bn=64 -> 4 M-tiles, 32 tiles total
  const int m0 = (tid / nt) * 16, n0 = (tid % nt) * 64;
  const int r = lane & 15, hi = lane >> 4;
  const float* A = Y + (size_t)m0 * Ln;
  v8f acc[4] = {};
  for (int k = 0; k < Ln; k += 4) {
    const float* ap = A + r * Ln + k + 2 * hi;
    v2f a; a.x = ap[0]; a.y = ap[1];
#pragma unroll
    for (int j = 0; j < 4; ++j) {
      const float* bp = linW + (n0 + j * 16 + r) * Ln + k + 2 * hi;  // B = linW^T
      v2f bv; bv.x = bp[0]; bv.y = bp[1];
      acc[j] = wmma4(a, bv, acc[j]);
    }
  }
#pragma unroll
  for (int j = 0; j < 4; ++j) {
    const int col = n0 + j * 16 + r;
    const float lb = linb[col];
#pragma unroll
    for (int i = 0; i < 8; ++i) {
      const int bb = m0 + i + 8 * hi;             // batch row
      const float v = acc[j][i] + lb;
      out[bb * Ln + col] = (wids[bb * Ln + col] != 0) ? v : -1e30f;
    }
  }
}

// ---------------------------------------------------------------- maskA
__global__ void k_maska(const int* __restrict__ y1, const int* __restrict__ y2,
                        float* __restrict__ out) {
  int t = blockIdx.x * blockDim.x + threadIdx.x;
  if (t >= Bn * MAXAn) return;
  int b = t / MAXAn, j = t % MAXAn;
  out[Bn * Ln + t] = (y1[b] + j <= y2[b]) ? 1.f : 0.f;
}

// ================================================================ launch
extern "C" void kernel_launch(void* const* d_in, const int* in_sizes, int n_in,
                              void* d_out, int out_size, void* d_ws, size_t ws_size,
                              hipStream_t stream) {
  const float* Semb  = (const float*)d_in[0];
  const int*   wids  = (const int*)d_in[1];
  const int*   y1    = (const int*)d_in[2];
  const int*   y2    = (const int*)d_in[3];
  const int*   edges = (const int*)d_in[4];
  const float* W1    = (const float*)d_in[5];
  const float* hwlW  = (const float*)d_in[6];
  const float* hwlB  = (const float*)d_in[7];
  const float* hwgW  = (const float*)d_in[8];
  const float* hwgB  = (const float*)d_in[9];
  const float* gcnW  = (const float*)d_in[10];
  const float* gcnB  = (const float*)d_in[11];
  const float* convW = (const float*)d_in[12];
  const float* convB = (const float*)d_in[13];
  const float* linW  = (const float*)d_in[14];
  const float* linB  = (const float*)d_in[15];
  float* out = (float*)d_out;
  float* ws  = (float*)d_ws;

  // workspace layout (floats)
  float* X0    = ws;                      // 8,388,608  [B,H,L] / [B,L,H]
  float* X1    = ws + 8388608;            // 8,388,608
  float* X2    = ws + 16777216;           // 8,388,608
  float* ADJ   = ws + 25165824;           // 16,777,216 [B,L,L]
  float* YPART = ws + 41943040;           // 32,768
  float* DEN   = ws + 41975808;           // 32,768
  float* YREL  = ws + 42008576;           // 32,768

  dim3 blk(256);

  // zero adj + ypart (contiguous)
  k_zero<<<dim3(4096), blk, 0, stream>>>(ADJ, 16777216 + 32768);

  // x = W1 * Semb
  k_conv1<<<dim3(16, 1, Bn), blk, 0, stream>>>(W1, Semb, X0);
  // highway x2
  k_highway<<<dim3(16, 1, Bn), blk, 0, stream>>>(hwgW, hwgB, hwlW, hwlB, X0, X1);
  k_highway<<<dim3(16, 1, Bn), blk, 0, stream>>>(hwgW + Hn * Hn, hwgB + Hn,
                                                 hwlW + Hn * Hn, hwlB + Hn, X1, X0);
  // adjacency + denom
  k_scatter<<<dim3((Bn * En + 255) / 256), blk, 0, stream>>>(edges, ADJ);
  k_denom<<<dim3((Bn * Ln + 255) / 256), blk, 0, stream>>>(ADJ, DEN);
  // h = x^T
  k_transpose<<<dim3(Ln / 32, Hn / 32, Bn), dim3(32, 8), 0, stream>>>(X0, X1);

  // GCN layer 0: X1 -> X0
  k_axh <<<dim3(16, 1, Bn), blk, 0, stream>>>(ADJ, X1, X2);
  k_gcnw<<<dim3(16, 1, Bn), blk, 0, stream>>>(X2, gcnW, gcnB, DEN, X0);
  k_yacc<<<dim3(128), blk, 0, stream>>>(X0, convW, YPART);
  // GCN layer 1: X0 -> X1
  k_axh <<<dim3(16, 1, Bn), blk, 0, stream>>>(ADJ, X0, X2);
  k_gcnw<<<dim3(16, 1, Bn), blk, 0, stream>>>(X2, gcnW + Hn * Hn, gcnB + Hn, DEN, X1);
  k_yacc<<<dim3(128), blk, 0, stream>>>(X1, convW + Hn, YPART);
  // GCN layer 2: X1 -> X0
  k_axh <<<dim3(16, 1, Bn), blk, 0, stream>>>(ADJ, X1, X2);
  k_gcnw<<<dim3(16, 1, Bn), blk, 0, stream>>>(X2, gcnW + 2 * Hn * Hn, gcnB + 2 * Hn, DEN, X0);
  k_yacc<<<dim3(128), blk, 0, stream>>>(X0, convW + 2 * Hn, YPART);

  // y = relu(ypart + conv_b); logits = mask(y * linW^T + lin_b)
  k_yrelu<<<dim3(128), blk, 0, stream>>>(YPART, convB, YREL);
  k_logits<<<dim3(4), blk, 0, stream>>>(YREL, linW, linB, wids, out);
  k_maska<<<dim3((Bn * MAXAn + 255) / 256), blk, 0, stream>>>(y1, y2, out);
}